// SAGEConvAgg_18580028522747
// MI455X (gfx1250) — compile-verified
//
#include <hip/hip_runtime.h>

#define N_NODES 100000
#define N_EDGES 1600000
#define D_FEAT  128

// ---------------------------------------------------------------------------
// Kernel 1: zero the accumulator (d_out) and the degree array (in d_ws).
// Harness poisons both with 0xAA before timing, so re-init every call.
// float4 stores -> global_store_b128, fully coalesced.
// ---------------------------------------------------------------------------
__global__ __launch_bounds__(256) void sage_zero_kernel(float4* __restrict__ out4,
                                                        unsigned* __restrict__ deg) {
    const size_t idx    = (size_t)blockIdx.x * blockDim.x + threadIdx.x;
    const size_t total4 = (size_t)N_NODES * (D_FEAT / 4);   // 3.2M float4
    if (idx < total4) {
        out4[idx] = make_float4(0.f, 0.f, 0.f, 0.f);
    }
    if (idx < N_NODES) {
        deg[idx] = 0u;
    }
}

// ---------------------------------------------------------------------------
// Kernel 2: persistent-wave edge scatter.
// One wave32 owns a grid-stride slice of the edge list. Per edge:
//   - lane i gathers bytes [16i,16i+16) of feat[src] with one global_load_b128
//     (512B per wave, coalesced; the 51.2MB feature table is L2-resident on
//     MI455X's 192MB L2, so gathers run at L2 bandwidth)
//   - 4 non-returning global_atomic_add_f32 into out[dst] (RMW at L2 atomic
//     units, tracked on STOREcnt -> fire-and-forget)
//   - lane 0 bumps the integer in-degree (global_atomic_add_u32,
//     order-invariant => deterministic)
// Software pipeline: while the current edge's atomics drain, the wave loads
// the NEXT edge's (src,dst) and issues global_prefetch_b8 on the next feature
// row, hiding the gather latency of the following iteration.
// ---------------------------------------------------------------------------
__global__ __launch_bounds__(256) void sage_scatter_kernel(const float* __restrict__ feat,
                                                           const int*   __restrict__ src,
                                                           const int*   __restrict__ dst,
                                                           float*       __restrict__ out,
                                                           unsigned*    __restrict__ deg) {
    const int lane    = threadIdx.x & 31;
    const int wid     = blockIdx.x * (blockDim.x >> 5) + (threadIdx.x >> 5);
    const int n_waves = gridDim.x * (blockDim.x >> 5);

    int e = wid;
    if (e >= N_EDGES) return;

    int s = src[e];
    int d = dst[e];

    for (;;) {
        // gather this lane's 16B slice of the source feature row
        const float4 v = ((const float4*)(feat + (size_t)s * D_FEAT))[lane];

        // pipeline: fetch next edge's indices + prefetch its feature row
        const int  en   = e + n_waves;
        const bool more = (en < N_EDGES);
        int sn = 0, dn = 0;
        if (more) {
            sn = src[en];
            dn = dst[en];
            __builtin_prefetch(feat + (size_t)sn * D_FEAT + (size_t)lane * 4, 0, 0);
        }

        float* orow = out + (size_t)d * D_FEAT + (size_t)lane * 4;
        atomicAdd(orow + 0, v.x);   // global_atomic_add_f32, no return
        atomicAdd(orow + 1, v.y);
        atomicAdd(orow + 2, v.z);
        atomicAdd(orow + 3, v.w);

        if (lane == 0) {
            atomicAdd(deg + d, 1u); // global_atomic_add_u32, deterministic
        }

        if (!more) break;
        e = en; s = sn; d = dn;
    }
}

// ---------------------------------------------------------------------------
// Kernel 3: normalize by clamped in-degree (DGL mean semantics: isolated
// nodes yield 0, achieved by dividing the zero-sum by max(deg,1)).
// float4 in/out -> b128 traffic.
// ---------------------------------------------------------------------------
__global__ __launch_bounds__(256) void sage_norm_kernel(float4* __restrict__ out4,
                                                        const unsigned* __restrict__ deg) {
    const size_t idx    = (size_t)blockIdx.x * blockDim.x + threadIdx.x;
    const size_t total4 = (size_t)N_NODES * (D_FEAT / 4);
    if (idx >= total4) return;

    const int   node = (int)(idx / (D_FEAT / 4));
    const float dg   = (float)deg[node];
    const float inv  = 1.0f / fmaxf(dg, 1.0f);

    float4 v = out4[idx];
    v.x *= inv; v.y *= inv; v.z *= inv; v.w *= inv;
    out4[idx] = v;
}

// ---------------------------------------------------------------------------
// Launch: inputs (feat f32 [100000*128], src i32 [1.6M], dst i32 [1.6M]);
// output f32 [100000*128]. Degree scratch (400KB, uint) lives in d_ws.
// ---------------------------------------------------------------------------
extern "C" void kernel_launch(void* const* d_in, const int* in_sizes, int n_in,
                              void* d_out, int out_size, void* d_ws, size_t ws_size,
                              hipStream_t stream) {
    const float* feat = (const float*)d_in[0];
    const int*   src  = (const int*)d_in[1];
    const int*   dst  = (const int*)d_in[2];
    float*       out  = (float*)d_out;
    unsigned*    deg  = (unsigned*)d_ws;       // N_NODES uints = 400 KB scratch

    const size_t total4 = (size_t)N_NODES * (D_FEAT / 4);     // 3,200,000
    const int    zblk   = 256;
    const int    zgrid  = (int)((total4 + zblk - 1) / zblk);  // 12,500 blocks

    sage_zero_kernel<<<zgrid, zblk, 0, stream>>>((float4*)out, deg);

    // Persistent scatter: 8192 blocks x 8 waves = 65,536 waves,
    // ~24 edges per wave in the grid-stride pipeline.
    sage_scatter_kernel<<<8192, 256, 0, stream>>>(feat, src, dst, out, deg);

    sage_norm_kernel<<<zgrid, zblk, 0, stream>>>((float4*)out, deg);
}